// StratifiedTraitDetector_63256278335892
// MI455X (gfx1250) — compile-verified
//
#include <hip/hip_runtime.h>
#include <hip/hip_bf16.h>
#include <math.h>

// Problem constants (from reference)
#define B_ROWS 16384
#define NSP    17396
#define NSP4   4349      // NSP / 4 (exact)
#define NC     6
#define TOPK   10
#define TF     64
#define NET    50
#define HID    25
#define CLD    52        // padded LDS leading dim for the 50-wide concat

typedef __attribute__((ext_vector_type(2))) float v2f;
typedef __attribute__((ext_vector_type(8))) float v8f;

// ---------------------------------------------------------------------------
// Kernel 1: streaming top-k (k=10) over 17396 f32 per row + trait gather.
// One wave32 per row; float4 coalesced loads; per-lane sorted top-10 in
// registers; 10-round wave argmax merge via lane shuffles; lanes 0..5 gather
// species_tensor rows (hot in L2) and write the species part of the output.
// ---------------------------------------------------------------------------
__device__ __forceinline__ void consider(float v, int i, float (&val)[TOPK], int (&idx)[TOPK]) {
    if (v > val[TOPK - 1]) {
        float cv = v; int ci = i;
#pragma unroll
        for (int j = 0; j < TOPK; ++j) {
            if (cv > val[j]) {
                float tv = val[j]; val[j] = cv; cv = tv;
                int   ti = idx[j]; idx[j] = ci; ci = ti;
            }
        }
    }
}

__global__ __launch_bounds__(256) void topk_species_kernel(
    const float* __restrict__ sp, const float* __restrict__ species,
    const float* __restrict__ tkW, float* __restrict__ out)
{
    const int lane = threadIdx.x & 31;
    const int row  = blockIdx.x * 8 + (threadIdx.x >> 5);
    const float4* base4 = (const float4*)(sp + (size_t)row * NSP);

    float val[TOPK]; int idx[TOPK];
#pragma unroll
    for (int j = 0; j < TOPK; ++j) { val[j] = -INFINITY; idx[j] = 0; }

    // 4349 float4 per row; 32 lanes/iter -> 136 iterations
    for (int it = 0; it < 136; ++it) {
        int q = it * 32 + lane;
        if (q < NSP4) {
            float4 v4 = base4[q];
            int e = q * 4;
            consider(v4.x, e + 0, val, idx);
            consider(v4.y, e + 1, val, idx);
            consider(v4.z, e + 2, val, idx);
            consider(v4.w, e + 3, val, idx);
        }
    }

    // Merge 32 sorted lists: 10 rounds of wave argmax, pop winner's head.
    float acc = 0.0f;
    for (int k = 0; k < TOPK; ++k) {
        float bv = val[0]; int bl = lane; int bi = idx[0];
#pragma unroll
        for (int off = 16; off >= 1; off >>= 1) {
            float ov = __shfl_xor(bv, off, 32);
            int   ol = __shfl_xor(bl, off, 32);
            int   oi = __shfl_xor(bi, off, 32);
            bool take = (ov > bv) || ((ov == bv) && (ol < bl));
            bv = take ? ov : bv; bl = take ? ol : bl; bi = take ? oi : bi;
        }
        if (lane == bl) {           // pop head of winning lane's list
#pragma unroll
            for (int j = 0; j < TOPK - 1; ++j) { val[j] = val[j + 1]; idx[j] = idx[j + 1]; }
            val[TOPK - 1] = -INFINITY;
        }
        if (lane < NC)
            acc += species[(size_t)bi * NC + lane] * tkW[k * NC + lane];
    }
    if (lane < NC)
        out[(size_t)row * NC + lane] = acc;
}

// ---------------------------------------------------------------------------
// Kernel 2: MLP chain with V_WMMA_F32_16X16X4_F32 (exact f32, matches ref).
// One wave per 16-row tile. Fragment layouts per ISA 7.12.2:
//   A (16x4 f32): lane m=L&15, half=L>>4; v0 = A[m][k0+2*half], v1 = +1
//   B (4x16 f32): v0 = W[k0+2*half][n0+(L&15)], v1 = next k row
//   C/D: VGPR i -> row (i + 8*half), col (L&15)
// Zero-fill keeps EXEC uniform through every WMMA.
// ---------------------------------------------------------------------------
__device__ __forceinline__ float gelu_exact(float x) {
    return 0.5f * x * (1.0f + erff(x * 0.70710678118654752f));
}
__device__ __forceinline__ float lrelu(float x) { return x > 0.0f ? x : 0.01f * x; }

__device__ __forceinline__ v8f wmma_gemm16(const float* A, int lda, int K,
                                           const float* W, int ldw, int n0, int N,
                                           int lane, int half)
{
    const int m  = lane & 15;
    const int n  = n0 + m;
    const bool nv = (n < N);
    v8f acc = {0.f, 0.f, 0.f, 0.f, 0.f, 0.f, 0.f, 0.f};
    for (int k0 = 0; k0 < K; k0 += 4) {
        const int ka = k0 + 2 * half;
        v2f a, b;
        a.x = (ka     < K)       ? A[m * lda + ka]        : 0.f;
        a.y = (ka + 1 < K)       ? A[m * lda + ka + 1]    : 0.f;
        b.x = (ka     < K && nv) ? W[ ka      * ldw + n]  : 0.f;
        b.y = (ka + 1 < K && nv) ? W[(ka + 1) * ldw + n]  : 0.f;
        acc = __builtin_amdgcn_wmma_f32_16x16x4_f32(
                  false, a, false, b, (short)0, acc, false, false);
    }
    return acc;
}

template <int D, int OFF>
__device__ __forceinline__ void process_group(
    const float* __restrict__ xg_rowbase,
    const float* __restrict__ w1, const float* __restrict__ b1,
    const float* __restrict__ w2, const float* __restrict__ b2,
    float* hbuf, float* cbuf, int lane, int half)
{
    const int col = lane & 15;
    // layer 1: [16 x D] @ [D x 64] + b1 -> exact GELU -> hbuf (ld=64)
#pragma unroll
    for (int nt = 0; nt < 4; ++nt) {
        v8f acc = wmma_gemm16(xg_rowbase, D, D, w1, TF, nt * 16, TF, lane, half);
        const int c = nt * 16 + col;
        const float bias = b1[c];
#pragma unroll
        for (int i = 0; i < 8; ++i)
            hbuf[(i + 8 * half) * TF + c] = gelu_exact(acc[i] + bias);
    }
    // layer 2: [16 x 64] @ [64 x D] + b2 -> exact GELU -> cat columns [OFF, OFF+D)
    constexpr int NT2 = (D + 15) / 16;
#pragma unroll
    for (int nt = 0; nt < NT2; ++nt) {
        v8f acc = wmma_gemm16(hbuf, TF, TF, w2, D, nt * 16, D, lane, half);
        const int c = nt * 16 + col;
        const bool cv = (c < D);
        const float bias = cv ? b2[c] : 0.f;
#pragma unroll
        for (int i = 0; i < 8; ++i) {
            float v = gelu_exact(acc[i] + bias);
            if (cv) cbuf[(i + 8 * half) * CLD + OFF + c] = v;
        }
    }
}

__global__ __launch_bounds__(128) void mlp_wmma_kernel(
    const float* __restrict__ x_soil, const float* __restrict__ x_climate,
    const float* __restrict__ x_veg,  const float* __restrict__ x_geo,
    const float* __restrict__ soil_w1, const float* __restrict__ soil_b1,
    const float* __restrict__ soil_w2, const float* __restrict__ soil_b2,
    const float* __restrict__ climate_w1, const float* __restrict__ climate_b1,
    const float* __restrict__ climate_w2, const float* __restrict__ climate_b2,
    const float* __restrict__ veg_w1, const float* __restrict__ veg_b1,
    const float* __restrict__ veg_w2, const float* __restrict__ veg_b2,
    const float* __restrict__ geo_w1, const float* __restrict__ geo_b1,
    const float* __restrict__ geo_w2, const float* __restrict__ geo_b2,
    const float* __restrict__ mf_w1, const float* __restrict__ mf_b1,
    const float* __restrict__ mf_w2, const float* __restrict__ mf_b2,
    float* __restrict__ out)
{
    __shared__ float hbuf_s[4][16 * TF];   // per-wave layer-1 / hidden staging
    __shared__ float cbuf_s[4][16 * CLD];  // per-wave concat staging (50 + pad)

    const int lane = threadIdx.x & 31;
    const int half = lane >> 4;
    const int wv   = threadIdx.x >> 5;
    const int r0   = (blockIdx.x * 4 + wv) * 16;
    float* hbuf = hbuf_s[wv];
    float* cbuf = cbuf_s[wv];

    process_group<20, 0 >(x_soil    + (size_t)r0 * 20, soil_w1,    soil_b1,    soil_w2,    soil_b2,    hbuf, cbuf, lane, half);
    process_group<15, 20>(x_climate + (size_t)r0 * 15, climate_w1, climate_b1, climate_w2, climate_b2, hbuf, cbuf, lane, half);
    process_group<10, 35>(x_veg     + (size_t)r0 * 10, veg_w1,     veg_b1,     veg_w2,     veg_b2,     hbuf, cbuf, lane, half);
    process_group<5,  45>(x_geo     + (size_t)r0 * 5,  geo_w1,     geo_b1,     geo_w2,     geo_b2,     hbuf, cbuf, lane, half);

    const int col = lane & 15;
    // merge layer 1: [16 x 50] @ [50 x 25] + b -> leaky relu -> hbuf cols 0..24
#pragma unroll
    for (int nt = 0; nt < 2; ++nt) {
        v8f acc = wmma_gemm16(cbuf, CLD, NET, mf_w1, HID, nt * 16, HID, lane, half);
        const int c = nt * 16 + col;
        const bool cv = (c < HID);
        const float bias = cv ? mf_b1[c] : 0.f;
#pragma unroll
        for (int i = 0; i < 8; ++i) {
            float v = lrelu(acc[i] + bias);
            if (cv) hbuf[(i + 8 * half) * TF + c] = v;
        }
    }
    // merge layer 2: [16 x 25] @ [25 x 6] + b -> leaky relu -> out +=
    {
        v8f acc = wmma_gemm16(hbuf, TF, HID, mf_w2, NC, 0, NC, lane, half);
        if (col < NC) {
            const float bias = mf_b2[col];
#pragma unroll
            for (int i = 0; i < 8; ++i) {
                const int r = r0 + i + 8 * half;
                out[(size_t)r * NC + col] += lrelu(acc[i] + bias);
            }
        }
    }
}

// ---------------------------------------------------------------------------
extern "C" void kernel_launch(void* const* d_in, const int* in_sizes, int n_in,
                              void* d_out, int out_size, void* d_ws, size_t ws_size,
                              hipStream_t stream)
{
    (void)in_sizes; (void)n_in; (void)d_ws; (void)ws_size; (void)out_size;

    const float* sp        = (const float*)d_in[0];
    const float* x_soil    = (const float*)d_in[1];
    const float* x_climate = (const float*)d_in[2];
    const float* x_veg     = (const float*)d_in[3];
    const float* x_geo     = (const float*)d_in[4];
    const float* species   = (const float*)d_in[5];
    const float* tkW       = (const float*)d_in[6];
    const float* soil_w1    = (const float*)d_in[7];
    const float* soil_b1    = (const float*)d_in[8];
    const float* soil_w2    = (const float*)d_in[9];
    const float* soil_b2    = (const float*)d_in[10];
    const float* climate_w1 = (const float*)d_in[11];
    const float* climate_b1 = (const float*)d_in[12];
    const float* climate_w2 = (const float*)d_in[13];
    const float* climate_b2 = (const float*)d_in[14];
    const float* veg_w1     = (const float*)d_in[15];
    const float* veg_b1     = (const float*)d_in[16];
    const float* veg_w2     = (const float*)d_in[17];
    const float* veg_b2     = (const float*)d_in[18];
    const float* geo_w1     = (const float*)d_in[19];
    const float* geo_b1     = (const float*)d_in[20];
    const float* geo_w2     = (const float*)d_in[21];
    const float* geo_b2     = (const float*)d_in[22];
    const float* mf_w1      = (const float*)d_in[23];
    const float* mf_b1      = (const float*)d_in[24];
    const float* mf_w2      = (const float*)d_in[25];
    const float* mf_b2      = (const float*)d_in[26];
    float* out = (float*)d_out;

    // 1) top-k scan + species trait sum (writes out)
    topk_species_kernel<<<B_ROWS / 8, 256, 0, stream>>>(sp, species, tkW, out);

    // 2) WMMA MLP chain (accumulates into out)
    mlp_wmma_kernel<<<B_ROWS / 64, 128, 0, stream>>>(
        x_soil, x_climate, x_veg, x_geo,
        soil_w1, soil_b1, soil_w2, soil_b2,
        climate_w1, climate_b1, climate_w2, climate_b2,
        veg_w1, veg_b1, veg_w2, veg_b2,
        geo_w1, geo_b1, geo_w2, geo_b2,
        mf_w1, mf_b1, mf_w2, mf_b2, out);
}